// DeuantizationLayer_1580547965018
// MI455X (gfx1250) — compile-verified
//
#include <hip/hip_runtime.h>

// D = A(16x4) x B(4x16) + C  via V_WMMA_F32_16X16X4_F32 (wave32).
// A = weight rows [8,4,2,1] (every row identical), B = bits of 16 symbols
// (one symbol per column), C = 0.5 broadcast. Every row of D holds the 16
// outputs; d[0] gives row0 (lanes 0-15) / row8 (lanes 16-31), identical values.
typedef float v2f __attribute__((ext_vector_type(2)));
typedef float v8f __attribute__((ext_vector_type(8)));

__global__ __launch_bounds__(256) void bit2num_wmma_kernel(
    const float* __restrict__ x, float* __restrict__ out,
    long long npairs, long long total)
{
    const int lane = threadIdx.x & 31;
    const int h    = lane >> 4;   // 0 -> holds K=0,1 ; 1 -> holds K=2,3
    const int n    = lane & 15;   // B-matrix column (symbol within tile)

    const long long wid    = (((long long)blockIdx.x * blockDim.x) + threadIdx.x) >> 5;
    const long long nwaves = (((long long)gridDim.x * blockDim.x)) >> 5;

    // A matrix (16x4 f32, 2 VGPRs): VGPR0 = K0 (lanes 0-15) / K2 (lanes 16-31),
    // VGPR1 = K1 / K3. Weights 2^(3-k) = {8,4,2,1}.
    v2f a;
    a.x = h ? 2.0f : 8.0f;
    a.y = h ? 1.0f : 4.0f;

    // Per-lane byte position of this lane's two bits within a 16-symbol tile.
    const int boff = n * 4 + 2 * h;   // in floats

    for (long long p = wid; p < npairs; p += nwaves) {
        const float* base = x + p * 128;   // two tiles = 32 symbols * 4 bits

        // B matrices: 8B non-temporal loads, each covering one tile's 256B.
        v2f b_e = __builtin_nontemporal_load((const v2f*)(base + boff));
        v2f b_o = __builtin_nontemporal_load((const v2f*)(base + 64 + boff));

        v8f c_e = {0.5f, 0.5f, 0.5f, 0.5f, 0.5f, 0.5f, 0.5f, 0.5f};
        v8f c_o = {0.5f, 0.5f, 0.5f, 0.5f, 0.5f, 0.5f, 0.5f, 0.5f};

        // 8 args: (neg_a, A, neg_b, B, c_mod, C, reuse_a, reuse_b)
        c_e = __builtin_amdgcn_wmma_f32_16x16x4_f32(
                  false, a, false, b_e, (short)0, c_e, false, false);
        c_o = __builtin_amdgcn_wmma_f32_16x16x4_f32(
                  false, a, false, b_o, (short)0, c_o, false, false);

        // Lanes 0-15 take even-tile outputs (D row 0), lanes 16-31 take
        // odd-tile outputs (D row 8, identical values). One coalesced 128B store.
        float v = (h == 0) ? c_e[0] : c_o[0];
        __builtin_nontemporal_store(v * 0.0625f, out + p * 32 + lane);
    }

    // Scalar tail for out_size not divisible by 32 (empty for reference shapes).
    for (long long r = npairs * 32 + ((long long)blockIdx.x * blockDim.x) + threadIdx.x;
         r < total; r += (long long)gridDim.x * blockDim.x) {
        const float* px = x + r * 4;
        float num = 8.0f * px[0] + 4.0f * px[1] + 2.0f * px[2] + px[3];
        out[r] = (num + 0.5f) * 0.0625f;
    }
}

extern "C" void kernel_launch(void* const* d_in, const int* in_sizes, int n_in,
                              void* d_out, int out_size, void* d_ws, size_t ws_size,
                              hipStream_t stream)
{
    const float* x   = (const float*)d_in[0];   // bits as f32 {0,1}
    float*       out = (float*)d_out;           // [batch, n_sym] f32
    (void)d_ws; (void)ws_size; (void)n_in; (void)in_sizes; // B hardcoded = 4

    const long long total  = (long long)out_size;
    const long long npairs = total / 32;        // 2 WMMA tiles (32 outputs) per iter

    const int threads = 256;                    // 8 wave32 per block
    const int blocks  = 2048;                   // 16384 waves, grid-stride loop

    bit2num_wmma_kernel<<<blocks, threads, 0, stream>>>(x, out, npairs, total);
}